// DKVMN_3487513444383
// MI455X (gfx1250) — compile-verified
//
#include <hip/hip_runtime.h>
#include <hip/hip_bf16.h>
#include <math.h>

#define NUM_CC 1000
#define DIM_D  128
#define MEM_M  64
#define BATCH  64
#define SEQ_L  512
#define NROWS  (BATCH * SEQ_L)   // 32768

typedef __attribute__((ext_vector_type(2))) float v2f;
typedef __attribute__((ext_vector_type(8))) float v8f;

__device__ __forceinline__ float sig_(float x) { return 1.0f / (1.0f + __expf(-x)); }

// ---------------------------------------------------------------------------
// K1: embedding gathers  k = Ek[q], v = Ev[q + 1000*r]
// ---------------------------------------------------------------------------
__global__ void __launch_bounds__(256)
k_gather(const int* __restrict__ q, const int* __restrict__ r,
         const float* __restrict__ Ek, const float* __restrict__ Ev,
         float* __restrict__ kout, float* __restrict__ vout) {
    int idx = blockIdx.x * 256 + threadIdx.x;     // over NROWS*128
    int i = idx >> 7;
    int d = idx & 127;
    int qi = q[i];
    int xi = qi + NUM_CC * r[i];
    kout[idx] = Ek[qi * DIM_D + d];
    vout[idx] = Ev[xi * DIM_D + d];
}

// ---------------------------------------------------------------------------
// K2: logits = k @ Mk^T  (16 rows/WG, 4 waves * 16 cols = 64 slots), softmax
// ---------------------------------------------------------------------------
__global__ void __launch_bounds__(128)
k_logits_softmax(const float* __restrict__ K, const float* __restrict__ Mk,
                 float* __restrict__ W) {
    __shared__ float lds_a[16 * 132];   // padded stride: conflict-free A frags
    __shared__ float lds_l[16 * 68];
    const int tid  = threadIdx.x;
    const int row0 = blockIdx.x * 16;

    for (int j = tid; j < 16 * 128; j += 128) {
        int rr = j >> 7, cc = j & 127;
        lds_a[rr * 132 + cc] = K[(row0 + rr) * DIM_D + cc];
    }
    __syncthreads();

    const int wave = tid >> 5, lane = tid & 31;
    const int half = lane >> 4, l16 = lane & 15;
    const int ncol = wave * 16 + l16;           // memory-slot index 0..63

    v8f acc = {};
    for (int s = 0; s < 32; ++s) {
        int kb = 4 * s + 2 * half;
        v2f af, bfrag;
        af.x = lds_a[l16 * 132 + kb];
        af.y = lds_a[l16 * 132 + kb + 1];
        // B[kk, m] = Mk[m*128 + kk] (k @ Mk^T) -> consecutive pair per lane
        bfrag.x = Mk[ncol * DIM_D + kb];
        bfrag.y = Mk[ncol * DIM_D + kb + 1];
        acc = __builtin_amdgcn_wmma_f32_16x16x4_f32(false, af, false, bfrag,
                                                    (short)0, acc, false, false);
    }
    for (int g = 0; g < 8; ++g) {
        int rr = g + 8 * half;
        lds_l[rr * 68 + ncol] = acc[g];
    }
    __syncthreads();

    if (tid < 16) {
        float mx = -1e30f;
        for (int m = 0; m < MEM_M; ++m) mx = fmaxf(mx, lds_l[tid * 68 + m]);
        float sum = 0.0f;
        for (int m = 0; m < MEM_M; ++m) sum += __expf(lds_l[tid * 68 + m] - mx);
        float inv = 1.0f / sum;
        for (int m = 0; m < MEM_M; ++m)
            W[(size_t)(row0 + tid) * MEM_M + m] = __expf(lds_l[tid * 68 + m] - mx) * inv;
    }
}

// ---------------------------------------------------------------------------
// K3: e = sigmoid(v@We + be), a = tanh(v@Wa + ba); A-fragment reused twice
// ---------------------------------------------------------------------------
__global__ void __launch_bounds__(256)
k_erase_add(const float* __restrict__ V,
            const float* __restrict__ We, const float* __restrict__ be,
            const float* __restrict__ Wa, const float* __restrict__ ba,
            float* __restrict__ E, float* __restrict__ A) {
    __shared__ float lds_a[16 * 132];
    const int tid  = threadIdx.x;
    const int row0 = blockIdx.x * 16;

    for (int j = tid; j < 16 * 128; j += 256) {
        int rr = j >> 7, cc = j & 127;
        lds_a[rr * 132 + cc] = V[(row0 + rr) * DIM_D + cc];
    }
    __syncthreads();

    const int wave = tid >> 5, lane = tid & 31;
    const int half = lane >> 4, l16 = lane & 15;
    const int ncol = wave * 16 + l16;           // output feature 0..127

    v8f ce = {}, ca = {};
    for (int s = 0; s < 32; ++s) {
        int kb = 4 * s + 2 * half;
        v2f af, bw, bv;
        af.x = lds_a[l16 * 132 + kb];
        af.y = lds_a[l16 * 132 + kb + 1];
        bw.x = We[kb * DIM_D + ncol];
        bw.y = We[(kb + 1) * DIM_D + ncol];
        bv.x = Wa[kb * DIM_D + ncol];
        bv.y = Wa[(kb + 1) * DIM_D + ncol];
        ce = __builtin_amdgcn_wmma_f32_16x16x4_f32(false, af, false, bw,
                                                   (short)0, ce, false, false);
        ca = __builtin_amdgcn_wmma_f32_16x16x4_f32(false, af, false, bv,
                                                   (short)0, ca, false, false);
    }
    float bev = be[ncol], bav = ba[ncol];
    for (int g = 0; g < 8; ++g) {
        size_t rr = (size_t)(row0 + g + 8 * half) * DIM_D + ncol;
        E[rr] = sig_(ce[g] + bev);
        A[rr] = tanhf(ca[g] + bav);
    }
}

// ---------------------------------------------------------------------------
// K4: sequential memory scan. One workgroup per batch element.
//     512 threads = (d 0..127) x (m-group 0..3); each thread keeps its
//     16-element Mv slice IN REGISTERS (no DS traffic in the hot loop).
//     Double-buffered w + partial-sum buffers -> one barrier per step.
// ---------------------------------------------------------------------------
__global__ void __launch_bounds__(512)
k_scan(const float* __restrict__ W, const float* __restrict__ E,
       const float* __restrict__ A, const float* __restrict__ Mv0,
       float* __restrict__ R) {
    __shared__ float wbuf[2][MEM_M];
    __shared__ float part[2][4][DIM_D];
    const int tid = threadIdx.x;
    const int d   = tid & 127;          // memory column
    const int mg  = tid >> 7;           // m-group 0..3
    const int m0  = mg * 16;
    const int b   = blockIdx.x;

    float mv[16];                       // this thread's Mv slice (registers)
    #pragma unroll
    for (int i = 0; i < 16; ++i) mv[i] = Mv0[(m0 + i) * DIM_D + d];

    const size_t base = (size_t)b * SEQ_L;
    if (tid < MEM_M) wbuf[0][tid] = W[base * MEM_M + tid];
    float ev = E[base * DIM_D + d];
    float av = A[base * DIM_D + d];
    __syncthreads();

    for (int t = 0; t < SEQ_L; ++t) {
        const int cur = t & 1;
        float evn = 0.0f, avn = 0.0f;
        if (t + 1 < SEQ_L) {                        // prefetch next step
            if (tid < MEM_M) wbuf[cur ^ 1][tid] = W[(base + t + 1) * MEM_M + tid];
            evn = E[(base + t + 1) * DIM_D + d];
            avn = A[(base + t + 1) * DIM_D + d];
        }
        float acc = 0.0f;
        #pragma unroll
        for (int i = 0; i < 16; ++i) {
            float wm = wbuf[cur][m0 + i];           // LDS broadcast
            acc = fmaf(wm, mv[i], acc);             // read BEFORE update
            mv[i] = mv[i] * (1.0f - wm * ev) + wm * av;
        }
        part[cur][mg][d] = acc;
        __syncthreads();
        if (mg == 0) {
            R[(base + t) * DIM_D + d] = part[cur][0][d] + part[cur][1][d] +
                                        part[cur][2][d] + part[cur][3][d];
        }
        ev = evn; av = avn;
    }
}

// ---------------------------------------------------------------------------
// K5: f = tanh([read||k] @ Wf + bf) ; p = sigmoid(f @ Wp + bp)
// ---------------------------------------------------------------------------
__global__ void __launch_bounds__(256)
k_final(const float* __restrict__ R, const float* __restrict__ K,
        const float* __restrict__ Wf, const float* __restrict__ bfv,
        const float* __restrict__ Wp, const float* __restrict__ bp,
        float* __restrict__ P) {
    __shared__ float lds_a[16 * 260];   // 16 x 256 concat tile, padded
    __shared__ float lds_f[16 * 132];
    const int tid  = threadIdx.x;
    const int row0 = blockIdx.x * 16;

    for (int j = tid; j < 16 * 256; j += 256) {
        int rr = j >> 8, cc = j & 255;
        float val = (cc < 128) ? R[(row0 + rr) * DIM_D + cc]
                               : K[(row0 + rr) * DIM_D + (cc - 128)];
        lds_a[rr * 260 + cc] = val;
    }
    __syncthreads();

    const int wave = tid >> 5, lane = tid & 31;
    const int half = lane >> 4, l16 = lane & 15;
    const int ncol = wave * 16 + l16;

    v8f cf = {};
    for (int s = 0; s < 64; ++s) {       // K = 256
        int kb = 4 * s + 2 * half;
        v2f af, bw;
        af.x = lds_a[l16 * 260 + kb];
        af.y = lds_a[l16 * 260 + kb + 1];
        bw.x = Wf[kb * DIM_D + ncol];
        bw.y = Wf[(kb + 1) * DIM_D + ncol];
        cf = __builtin_amdgcn_wmma_f32_16x16x4_f32(false, af, false, bw,
                                                   (short)0, cf, false, false);
    }
    float bfn = bfv[ncol];
    for (int g = 0; g < 8; ++g) {
        int rr = g + 8 * half;
        lds_f[rr * 132 + ncol] = tanhf(cf[g] + bfn);
    }
    __syncthreads();

    if (tid < 16) {
        float acc = bp[0];
        for (int d = 0; d < DIM_D; ++d)
            acc = fmaf(lds_f[tid * 132 + d], Wp[d], acc);
        P[row0 + tid] = sig_(acc);
    }
}

// ---------------------------------------------------------------------------
extern "C" void kernel_launch(void* const* d_in, const int* in_sizes, int n_in,
                              void* d_out, int out_size, void* d_ws, size_t ws_size,
                              hipStream_t stream) {
    const int*   q   = (const int*)d_in[0];
    const int*   r   = (const int*)d_in[1];
    const float* Ek  = (const float*)d_in[2];
    const float* Ev  = (const float*)d_in[3];
    const float* Mk  = (const float*)d_in[4];
    const float* Mv0 = (const float*)d_in[5];
    const float* We  = (const float*)d_in[6];
    const float* be  = (const float*)d_in[7];
    const float* Wa  = (const float*)d_in[8];
    const float* ba  = (const float*)d_in[9];
    const float* Wf  = (const float*)d_in[10];
    const float* bf  = (const float*)d_in[11];
    const float* Wp  = (const float*)d_in[12];
    const float* bp  = (const float*)d_in[13];
    float* P = (float*)d_out;

    float* ws = (float*)d_ws;
    float* kb = ws;                                  // [NROWS,128]
    float* vb = kb + (size_t)NROWS * DIM_D;          // [NROWS,128]
    float* wb = vb + (size_t)NROWS * DIM_D;          // [NROWS,64]
    float* eb = wb + (size_t)NROWS * MEM_M;          // [NROWS,128]
    float* ab = eb + (size_t)NROWS * DIM_D;          // [NROWS,128]
    float* rb = ab + (size_t)NROWS * DIM_D;          // [NROWS,128]

    k_gather        <<<(NROWS * DIM_D) / 256, 256, 0, stream>>>(q, r, Ek, Ev, kb, vb);
    k_logits_softmax<<<NROWS / 16, 128, 0, stream>>>(kb, Mk, wb);
    k_erase_add     <<<NROWS / 16, 256, 0, stream>>>(vb, We, be, Wa, ba, eb, ab);
    k_scan          <<<BATCH, 512, 0, stream>>>(wb, eb, ab, Mv0, rb);
    k_final         <<<NROWS / 16, 256, 0, stream>>>(rb, kb, Wf, bf, Wp, bp, P);

    (void)in_sizes; (void)n_in; (void)out_size; (void)ws_size;
}